// VolumeRenderingNeuS_89790586290721
// MI455X (gfx1250) — compile-verified
//
#include <hip/hip_runtime.h>
#include <hip/hip_bf16.h>
#include <math.h>

// NeuS volume rendering for MI455X (gfx1250, wave32).
// Bandwidth-bound (~0.7 GB total traffic -> ~30us at 23.3 TB/s).
// Block scans use V_WMMA_F32_16X16X4_F32: exclusive scan of a 16x16 matrix X is
//   Y = Lstrict*X*J + X*Ustrict   (strict triangular ones, J = all-ones)
// each 16x16x16 product = 4 chained K=4 WMMAs => 12 wmma per tile scan.

#define NT 256
#define ITEMS 8
#define TILE (NT * ITEMS)

typedef float v2f __attribute__((ext_vector_type(2)));
typedef float v8f __attribute__((ext_vector_type(8)));

// ---------------- small load/store helpers (b128 vector memops) ----------------
__device__ __forceinline__ void load8f(const float* __restrict__ p, float* v) {
  float4 a = ((const float4*)p)[0];
  float4 b = ((const float4*)p)[1];
  v[0] = a.x; v[1] = a.y; v[2] = a.z; v[3] = a.w;
  v[4] = b.x; v[5] = b.y; v[6] = b.z; v[7] = b.w;
}
__device__ __forceinline__ void store8f(float* __restrict__ p, const float* v) {
  float4 a = make_float4(v[0], v[1], v[2], v[3]);
  float4 b = make_float4(v[4], v[5], v[6], v[7]);
  ((float4*)p)[0] = a;
  ((float4*)p)[1] = b;
}
__device__ __forceinline__ void load8i(const int* __restrict__ p, int* v) {
  int4 a = ((const int4*)p)[0];
  int4 b = ((const int4*)p)[1];
  v[0] = a.x; v[1] = a.y; v[2] = a.z; v[3] = a.w;
  v[4] = b.x; v[5] = b.y; v[6] = b.z; v[7] = b.w;
}
__device__ __forceinline__ void load24f(const float* __restrict__ p, float* v) {
#pragma unroll
  for (int j = 0; j < 6; ++j) {
    float4 a = ((const float4*)p)[j];
    v[4 * j + 0] = a.x; v[4 * j + 1] = a.y; v[4 * j + 2] = a.z; v[4 * j + 3] = a.w;
  }
}

// ---------------- NeuS per-sample math ----------------
__device__ __forceinline__ float neus_log1malpha(float sd, float gx, float gy, float gz,
                                                 float dx, float dy, float dz,
                                                 float dtv, float car, float beta) {
  float tc = dx * gx + dy * gy + dz * gz;                 // true_cos
  float r1 = fmaxf(0.5f - 0.5f * tc, 0.f);
  float r2 = fmaxf(-tc, 0.f);
  float ic = -(r1 * (1.f - car) + r2 * car);              // iter_cos
  float e  = ic * dtv * 0.5f;
  float pn = (sd + e) * beta;                             // next sdf * beta
  float pp = (sd - e) * beta;                             // prev sdf * beta
  float ncdf = 1.f / (1.f + __expf(-pn));
  float pcdf = 1.f / (1.f + __expf(-pp));
  float alpha = (pcdf - ncdf + 1e-6f) / (pcdf + 1e-6f);
  alpha = fminf(fmaxf(alpha, 0.f), 1.f);
  return __logf(1.f - alpha + 1e-6f);                     // log(one_minus_alpha)
}

// ---------------- 256-element exclusive scan via f32 WMMA (wave 0 only) --------
// tot/pmat/esc are LDS. lane in [0,32). EXEC is all ones inside the tid<32 branch.
__device__ __forceinline__ void wmma_scan256(const float* tot, float* pmat, float* esc,
                                             int lane) {
  const int  halfRow = lane & 15;
  const int  koff    = (lane < 16) ? 0 : 2;   // which K pair this half-wave holds
  const int  mbase   = (lane < 16) ? 0 : 8;   // C/D row group

  // P = Lstrict * X   (4 chained 16x16x4 WMMAs)
  v8f accP = {0.f, 0.f, 0.f, 0.f, 0.f, 0.f, 0.f, 0.f};
#pragma unroll
  for (int k = 0; k < 4; ++k) {
    const int c0 = 4 * k + koff, c1 = c0 + 1;
    v2f a, b;
    a.x = (c0 < halfRow) ? 1.f : 0.f;          // Lstrict[halfRow][c0]
    a.y = (c1 < halfRow) ? 1.f : 0.f;
    b.x = tot[c0 * 16 + halfRow];              // X[c0][col=halfRow]
    b.y = tot[c1 * 16 + halfRow];
    accP = __builtin_amdgcn_wmma_f32_16x16x4_f32(false, a, false, b, (short)0, accP,
                                                 false, false);
  }
#pragma unroll
  for (int j = 0; j < 8; ++j) pmat[(mbase + j) * 16 + halfRow] = accP[j];
  // same-wave LDS store->load ordering (CDNA5 split counter)
  __asm volatile("s_wait_dscnt 0x0" ::: "memory");

  // Y = X * Ustrict + P * J
  v8f accY = {0.f, 0.f, 0.f, 0.f, 0.f, 0.f, 0.f, 0.f};
#pragma unroll
  for (int k = 0; k < 4; ++k) {
    const int c0 = 4 * k + koff, c1 = c0 + 1;
    v2f a, b;
    a.x = tot[halfRow * 16 + c0];              // X[row=halfRow][c0]
    a.y = tot[halfRow * 16 + c1];
    b.x = (c0 < halfRow) ? 1.f : 0.f;          // Ustrict[c0][col=halfRow]
    b.y = (c1 < halfRow) ? 1.f : 0.f;
    accY = __builtin_amdgcn_wmma_f32_16x16x4_f32(false, a, false, b, (short)0, accY,
                                                 false, false);
  }
#pragma unroll
  for (int k = 0; k < 4; ++k) {
    const int c0 = 4 * k + koff, c1 = c0 + 1;
    v2f a, b;
    a.x = pmat[halfRow * 16 + c0];             // P[row=halfRow][c0]
    a.y = pmat[halfRow * 16 + c1];
    b.x = 1.f;                                 // J (all ones)
    b.y = 1.f;
    accY = __builtin_amdgcn_wmma_f32_16x16x4_f32(false, a, false, b, (short)0, accY,
                                                 false, false);
  }
#pragma unroll
  for (int j = 0; j < 8; ++j) esc[(mbase + j) * 16 + halfRow] = accY[j];
}

// thread-total -> per-thread exclusive prefix within the tile
__device__ __forceinline__ float tile_thread_excl(float s_thread, float* tot, float* pmat,
                                                  float* esc, int tid) {
  tot[tid] = s_thread;
  __syncthreads();
  if (tid < 32) wmma_scan256(tot, pmat, esc, tid);
  __syncthreads();
  return esc[tid];
}

// load 8 log values for this thread (zeros past N)
__device__ __forceinline__ void load_lv(const float* lw, int base, int N, float* lv) {
  if (base + ITEMS <= N) {
    load8f(lw + base, lv);
  } else {
#pragma unroll
    for (int k = 0; k < ITEMS; ++k) lv[k] = (base + k < N) ? lw[base + k] : 0.f;
  }
}

// ---------------- K0: zero per-ray rgb output ----------------
__global__ void k0_zero(float* __restrict__ p, int n) {
  int i = blockIdx.x * blockDim.x + threadIdx.x;
  if (i < n) p[i] = 0.f;
}

// ---------------- K1: alpha/log + block sums ----------------
__global__ void k1_alpha_log(const float* __restrict__ sdf, const float* __restrict__ grad,
                             const float* __restrict__ dirs, const float* __restrict__ dt,
                             const float* __restrict__ carp, const float* __restrict__ betap,
                             float* __restrict__ lw, float* __restrict__ bsum, int N) {
  __shared__ float red[NT];
  const int tid  = threadIdx.x;
  const int base = blockIdx.x * TILE + tid * ITEMS;
  const float car = carp[0], beta = betap[0];

  float sd[ITEMS], dtv[ITEMS], g[3 * ITEMS], dv[3 * ITEMS];
  const bool full = (base + ITEMS) <= N;
  if (full) {
    load8f(sdf + base, sd);
    load8f(dt + base, dtv);
    load24f(grad + 3 * base, g);
    load24f(dirs + 3 * base, dv);
  } else {
#pragma unroll
    for (int k = 0; k < ITEMS; ++k) {
      const int i = base + k;
      const bool ok = i < N;
      sd[k]  = ok ? sdf[i] : 0.f;
      dtv[k] = ok ? dt[i] : 0.f;
#pragma unroll
      for (int c = 0; c < 3; ++c) {
        g[3 * k + c]  = ok ? grad[3 * i + c] : 0.f;
        dv[3 * k + c] = ok ? dirs[3 * i + c] : 0.f;
      }
    }
  }

  float lv[ITEMS];
  float s = 0.f;
#pragma unroll
  for (int k = 0; k < ITEMS; ++k) {
    float l = neus_log1malpha(sd[k], g[3 * k], g[3 * k + 1], g[3 * k + 2],
                              dv[3 * k], dv[3 * k + 1], dv[3 * k + 2],
                              dtv[k], car, beta);
    if (!full && (base + k) >= N) l = 0.f;
    lv[k] = l;
    s += l;
  }

  if (full) {
    store8f(lw + base, lv);
  } else {
#pragma unroll
    for (int k = 0; k < ITEMS; ++k)
      if (base + k < N) lw[base + k] = lv[k];
  }

  red[tid] = s;
  __syncthreads();
#pragma unroll
  for (int off = NT / 2; off > 0; off >>= 1) {
    if (tid < off) red[tid] += red[tid + off];
    __syncthreads();
  }
  if (tid == 0) bsum[blockIdx.x] = red[0];
}

// ---------------- K2: single-block exclusive scan of block sums ----------------
__global__ void k2_scan_bsums(const float* __restrict__ bsum, float* __restrict__ boff,
                              int nb) {
  __shared__ float sh[NT];
  const int tid = threadIdx.x;
  float carry = 0.f;
  for (int start = 0; start < nb; start += NT) {
    const int i = start + tid;
    const float v = (i < nb) ? bsum[i] : 0.f;
    sh[tid] = v;
    __syncthreads();
    float x = v;
    for (int off = 1; off < NT; off <<= 1) {
      float t = (tid >= off) ? sh[tid - off] : 0.f;
      __syncthreads();
      x += t;
      sh[tid] = x;
      __syncthreads();
    }
    if (i < nb) boff[i] = carry + x - v;   // exclusive
    float tot = sh[NT - 1];
    __syncthreads();
    carry += tot;
  }
}

// ---------------- K3: segment heads -> seg_base scatter ----------------
__global__ void k3_seg_heads(const float* __restrict__ lw, const int* __restrict__ rid,
                             const float* __restrict__ boff, float* __restrict__ seg_base,
                             int N) {
  __shared__ float tot[NT];
  __shared__ float pmat[NT];
  __shared__ float esc[NT];
  const int tid  = threadIdx.x;
  const int base = blockIdx.x * TILE + tid * ITEMS;

  float lv[ITEMS];
  load_lv(lw, base, N, lv);
  float incl[ITEMS];
  float s = 0.f;
#pragma unroll
  for (int k = 0; k < ITEMS; ++k) { s += lv[k]; incl[k] = s; }

  const float te = tile_thread_excl(s, tot, pmat, esc, tid);
  const float bb = boff[blockIdx.x] + te;

  int r[ITEMS];
  if (base + ITEMS <= N) {
    load8i(rid + base, r);
  } else {
#pragma unroll
    for (int k = 0; k < ITEMS; ++k) r[k] = (base + k < N) ? rid[base + k] : 0;
  }
  int prev = 0;
  if (base > 0 && base < N) prev = rid[base - 1];

#pragma unroll
  for (int k = 0; k < ITEMS; ++k) {
    const int i = base + k;
    if (i < N) {
      const int cur = r[k];
      const int pv  = k ? r[k - 1] : prev;
      if (i == 0 || cur != pv) {
        seg_base[cur] = bb + (k ? incl[k - 1] : 0.f);   // c_excl at segment head
      }
    }
  }
}

// ---------------- K4: weights + per-ray rgb accumulation ----------------
__global__ void k4_weights(const float* __restrict__ lw, const int* __restrict__ rid,
                           const float* __restrict__ boff, const float* __restrict__ rgb,
                           const float* __restrict__ seg_base, float* __restrict__ out_rgb,
                           float* __restrict__ out_w, int N) {
  __shared__ float tot[NT];
  __shared__ float pmat[NT];
  __shared__ float esc[NT];
  const int tid  = threadIdx.x;
  const int base = blockIdx.x * TILE + tid * ITEMS;

  float lv[ITEMS];
  load_lv(lw, base, N, lv);
  float incl[ITEMS];
  float s = 0.f;
#pragma unroll
  for (int k = 0; k < ITEMS; ++k) { s += lv[k]; incl[k] = s; }

  const float te = tile_thread_excl(s, tot, pmat, esc, tid);
  const float bb = boff[blockIdx.x] + te;

  const bool full = (base + ITEMS) <= N;
  int r[ITEMS];
  float rg[3 * ITEMS];
  if (full) {
    load8i(rid + base, r);
    load24f(rgb + 3 * base, rg);
  } else {
#pragma unroll
    for (int k = 0; k < ITEMS; ++k) {
      const int i = base + k;
      const bool ok = i < N;
      r[k] = ok ? rid[i] : 0;
#pragma unroll
      for (int c = 0; c < 3; ++c) rg[3 * k + c] = ok ? rgb[3 * i + c] : 0.f;
    }
  }

  float wbuf[ITEMS];
  float a0 = 0.f, a1 = 0.f, a2 = 0.f;
  int cur = r[0];
#pragma unroll
  for (int k = 0; k < ITEMS; ++k) {
    const int i = base + k;
    if (i < N) {
      const float ce    = bb + (k ? incl[k - 1] : 0.f);       // global log-cumsum (excl)
      const float om    = __expf(lv[k]);                      // 1 - alpha + 1e-6
      const float alpha = 1.f + 1e-6f - om;
      const float T     = __expf(ce - seg_base[r[k]]);        // transmittance
      const float w     = alpha * T;
      wbuf[k] = w;
      if (r[k] != cur) {                                       // run-length flush
        atomicAdd(&out_rgb[3 * cur + 0], a0);
        atomicAdd(&out_rgb[3 * cur + 1], a1);
        atomicAdd(&out_rgb[3 * cur + 2], a2);
        a0 = a1 = a2 = 0.f;
        cur = r[k];
      }
      a0 += w * rg[3 * k + 0];
      a1 += w * rg[3 * k + 1];
      a2 += w * rg[3 * k + 2];
    } else {
      wbuf[k] = 0.f;
    }
  }
  if (base < N) {
    atomicAdd(&out_rgb[3 * cur + 0], a0);
    atomicAdd(&out_rgb[3 * cur + 1], a1);
    atomicAdd(&out_rgb[3 * cur + 2], a2);
  }

  if (full) {
    store8f(out_w + base, wbuf);
  } else {
#pragma unroll
    for (int k = 0; k < ITEMS; ++k)
      if (base + k < N) out_w[base + k] = wbuf[k];
  }
}

// ---------------- host entry ----------------
extern "C" void kernel_launch(void* const* d_in, const int* in_sizes, int n_in,
                              void* d_out, int out_size, void* d_ws, size_t ws_size,
                              hipStream_t stream) {
  const int N = in_sizes[0];                 // n_samples
  const int nrays3 = out_size - N;           // n_rays * 3
  const int nb = (N + TILE - 1) / TILE;

  const float* sdf  = (const float*)d_in[0];
  const float* grad = (const float*)d_in[1];
  const float* dirs = (const float*)d_in[2];
  const float* dt   = (const float*)d_in[3];
  const float* rgb  = (const float*)d_in[4];
  const float* car  = (const float*)d_in[5];
  const float* beta = (const float*)d_in[6];
  const int*   rid  = (const int*)d_in[7];

  float* out_rgb = (float*)d_out;
  float* out_w   = (float*)d_out + nrays3;

  // workspace layout: l (N) | block sums (nb) | block offsets (nb) | seg_base (n_rays)
  float* lw   = (float*)d_ws;
  float* bsum = lw + N;
  float* boff = bsum + nb;
  float* segb = boff + nb;

  k0_zero<<<(nrays3 + NT - 1) / NT, NT, 0, stream>>>(out_rgb, nrays3);
  k1_alpha_log<<<nb, NT, 0, stream>>>(sdf, grad, dirs, dt, car, beta, lw, bsum, N);
  k2_scan_bsums<<<1, NT, 0, stream>>>(bsum, boff, nb);
  k3_seg_heads<<<nb, NT, 0, stream>>>(lw, rid, boff, segb, N);
  k4_weights<<<nb, NT, 0, stream>>>(lw, rid, boff, rgb, segb, out_rgb, out_w, N);
}